// MultiHeadNetWithNestedHistory_41910290874924
// MI455X (gfx1250) — compile-verified
//
#include <hip/hip_runtime.h>
#include <math.h>

#define Bq   16384
#define Gq   17
#define Kq   12
#define KDq  16
#define DKq  16
#define DMq  32
#define Tq   4
#define SDq  64
#define H1q  256
#define H2q  128
#define EPSq 1e-5f

typedef float v2f __attribute__((ext_vector_type(2)));
typedef float v8f __attribute__((ext_vector_type(8)));

#if defined(__gfx1250__) && \
    __has_builtin(__builtin_amdgcn_global_load_async_to_lds_b128) && \
    __has_builtin(__builtin_amdgcn_s_wait_asynccnt)
#define USE_ASYNC_LDS 1
typedef int v4i __attribute__((vector_size(4 * sizeof(int))));
typedef v4i __attribute__((address_space(1))) as1_v4i;   // global
typedef v4i __attribute__((address_space(3))) as3_v4i;   // LDS
#endif

__device__ __forceinline__ v8f wmma4(v2f a, v2f b, v8f c) {
  // V_WMMA_F32_16X16X4_F32 : D = A(16x4,f32) * B(4x16,f32) + C(16x16,f32)
  return __builtin_amdgcn_wmma_f32_16x16x4_f32(false, a, false, b, (short)0, c,
                                               false, false);
}

// ---------------------------------------------------------------------------
// Kernel 1: per-(b,g) kick encoder (WMMA) + inner attention pool + game encoder
// one wave32 per game; D-layout: lane holds col n=lane&15, rows r + 8*(lane>>4)
// ---------------------------------------------------------------------------
__global__ __launch_bounds__(256) void encode_games_kernel(
    const float* __restrict__ x_kicks, const float* __restrict__ W_ke,
    const float* __restrict__ b_ke, const float* __restrict__ q_inner,
    const float* __restrict__ W_ge, const float* __restrict__ b_ge,
    const float* __restrict__ pos_emb, const unsigned char* __restrict__ inner_mask,
    float* __restrict__ enc_out) {
  const int lane = threadIdx.x & 31;
  const long wave = (long)blockIdx.x * 8 + (threadIdx.x >> 5);
  if (wave >= (long)Bq * Gq) return;
  const int b = (int)(wave / Gq);
  const int g = (int)(wave % Gq);
  const int n = lane & 15;
  const int hi = lane >> 4;
  const float* Xk = x_kicks + ((long)b * Gq + g) * (Kq * KDq);

  // branch-free zero padding for rows 12..15: clamp address, scale by 0/1
  const int rowc = (n < Kq) ? n : 0;
  const float msk = (n < Kq) ? 1.f : 0.f;

  // kick_enc = relu(X(12x16,pad to 16) @ W_ke(16x16) + b_ke)
  v8f acc = {0.f, 0.f, 0.f, 0.f, 0.f, 0.f, 0.f, 0.f};
#pragma unroll
  for (int s = 0; s < 4; ++s) {
    int kk = s * 4 + hi * 2;
    v2f a = *(const v2f*)(Xk + rowc * KDq + kk);  // 8B aligned (kk even)
    a = a * msk;
    v2f bb;
    bb.x = W_ke[kk * DKq + n];
    bb.y = W_ke[(kk + 1) * DKq + n];
    acc = wmma4(a, bb, acc);
  }

  const float bias = b_ke[n];
  const float qn = q_inner[n];
  float ke[8], ai[8];
#pragma unroll
  for (int r = 0; r < 8; ++r) {
    float c = fmaxf(acc[r] + bias, 0.f);
    ke[r] = c;
    float t = c * qn * 0.25f;  // DK^-0.5 = 1/4
#pragma unroll
    for (int off = 8; off > 0; off >>= 1) t += __shfl_xor(t, off, 16);
    ai[r] = t;  // score for kick k = r + 8*hi (uniform within half)
  }

  const unsigned char* im = inner_mask + ((long)b * Gq + g) * Kq;
  float mx = -INFINITY;
  bool valid[8];
#pragma unroll
  for (int r = 0; r < 8; ++r) {
    int k = r + hi * 8;
    int kc = (k < Kq) ? k : 0;
    unsigned char mb = im[kc];           // unconditional byte load
    bool v = (bool)((k < Kq) & (mb != 0));
    valid[r] = v;
    if (v) mx = fmaxf(mx, ai[r]);
  }
  mx = fmaxf(mx, __shfl_xor(mx, 16, 32));
  float sum = 0.f, wk[8];
#pragma unroll
  for (int r = 0; r < 8; ++r) {
    float e = valid[r] ? __expf(ai[r] - mx) : 0.f;
    wk[r] = e;
    sum += e;
  }
  sum += __shfl_xor(sum, 16, 32);
  const float inv = sum > 0.f ? 1.f / sum : 0.f;
  float pg = 0.f;
#pragma unroll
  for (int r = 0; r < 8; ++r) pg += wk[r] * inv * ke[r];
  pg += __shfl_xor(pg, 16, 32);  // per_game[e = lane&15], replicated per half

  // game encoder: encoded[m=lane] = relu(per_game @ W_ge + b_ge) + pos_emb
  float a2 = b_ge[lane];
#pragma unroll
  for (int e = 0; e < 16; ++e) {
    float pe = __shfl(pg, e, 16);
    a2 += pe * W_ge[e * DMq + lane];
  }
  enc_out[((long)b * Gq + g) * DMq + lane] = fmaxf(a2, 0.f) + pos_emb[g * DMq + lane];
}

// ---------------------------------------------------------------------------
// Kernel 2: outer attention pool + LayerNorm + backbone + heads.
// 128 threads (4 waves) per block, 16 batch rows per block.
// ---------------------------------------------------------------------------
__global__ __launch_bounds__(128) void fused_tail_kernel(
    const float* __restrict__ x_static, const float* __restrict__ enc_in,
    const float* __restrict__ q_outer, const float* __restrict__ log_temp,
    const float* __restrict__ ln_g, const float* __restrict__ ln_b,
    const float* __restrict__ W1, const float* __restrict__ b1,
    const float* __restrict__ g1, const float* __restrict__ be1,
    const float* __restrict__ m1, const float* __restrict__ v1,
    const float* __restrict__ W2, const float* __restrict__ b2,
    const float* __restrict__ g2, const float* __restrict__ be2,
    const float* __restrict__ m2, const float* __restrict__ v2,
    const float* __restrict__ Wh1, const float* __restrict__ bh1,
    const float* __restrict__ Wh2, const float* __restrict__ bh2,
    const unsigned char* __restrict__ outer_mask, float* __restrict__ out) {
  __shared__ __align__(16) float s_enc[16 * 17 * 36];   // padded stride 36
  __shared__ float s_w[16 * 8 * 20];                    // softmax weights, pad 17->20
  __shared__ float s_hist[16 * 4 * 68];                 // LN'd history, pad 64->68
  __shared__ __align__(16) float s_xs[16 * 68];         // x_static tile, pad 64->68
  __shared__ float s_z1[16 * 260];                      // layer1 out, pad 256->260
  __shared__ float s_z2[16 * 132];                      // layer2 out, pad 128->132

  const int tid = threadIdx.x;
  const int lane = tid & 31;
  const int wv = tid >> 5;
  const int row0 = blockIdx.x * 16;
  const int n = lane & 15;
  const int hi = lane >> 4;

  // ---- cooperative loads into LDS (async-to-LDS path on CDNA5) ----
#ifdef USE_ASYNC_LDS
  for (int i = tid; i < (16 * SDq) / 4; i += 128) {      // 256 x b128
    int r = i >> 4, c = (i & 15) * 4;
    __builtin_amdgcn_global_load_async_to_lds_b128(
        (as1_v4i*)(x_static + (long)(row0 + r) * SDq + c),
        (as3_v4i*)&s_xs[r * 68 + c], 0, 0);
  }
  for (int i = tid; i < (16 * Gq * DMq) / 4; i += 128) { // 2176 x b128
    int flat = i * 4;
    int r = flat / (Gq * DMq);
    int rem = flat % (Gq * DMq);
    int g = rem >> 5, d = rem & 31;
    __builtin_amdgcn_global_load_async_to_lds_b128(
        (as1_v4i*)(enc_in + ((long)(row0 + r) * Gq + g) * DMq + d),
        (as3_v4i*)&s_enc[(r * Gq + g) * 36 + d], 0, 0);
  }
  __builtin_amdgcn_s_wait_asynccnt(0);
#else
  for (int i = tid; i < 16 * SDq; i += 128) {
    int r = i >> 6, c = i & 63;
    s_xs[r * 68 + c] = x_static[(long)(row0 + r) * SDq + c];
  }
  for (int i = tid; i < (16 * Gq * DMq) / 4; i += 128) {
    int flat = i * 4;
    int r = flat / (Gq * DMq);
    int rem = flat % (Gq * DMq);
    int g = rem >> 5, d = rem & 31;
    *(float4*)&s_enc[(r * Gq + g) * 36 + d] =
        *(const float4*)&enc_in[((long)(row0 + r) * Gq + g) * DMq + d];
  }
#endif
  __syncthreads();

  // ---- stage 1: outer attention + LayerNorm (4 rows per wave) ----
  const float inv_sqrt_dm = 0.17677669529663687f;  // 32^-0.5
  float scl[4];
#pragma unroll
  for (int t = 0; t < 4; ++t) scl[t] = inv_sqrt_dm * __expf(-log_temp[t]);

  for (int i = 0; i < 4; ++i) {
    int rb = wv * 4 + i;
    int b = row0 + rb;
    bool gv = lane < Gq;
    int gl = gv ? lane : 0;
    int om = (int)outer_mask[(long)b * Gq + gl] & (gv ? 1 : 0);
    float sc[8] = {0.f, 0.f, 0.f, 0.f, 0.f, 0.f, 0.f, 0.f};
    for (int d = 0; d < DMq; ++d) {
      float e = gv ? s_enc[(rb * Gq + lane) * 36 + d] : 0.f;
#pragma unroll
      for (int q = 0; q < 8; ++q) sc[q] += e * q_outer[q * DMq + d];
    }
#pragma unroll
    for (int q = 0; q < 8; ++q) {
      float s = sc[q] * scl[q >> 1];
      float sm = om ? s : -INFINITY;
#pragma unroll
      for (int off = 16; off > 0; off >>= 1) sm = fmaxf(sm, __shfl_xor(sm, off, 32));
      float ex = om ? __expf(s - sm) : 0.f;
      float ss = ex;
#pragma unroll
      for (int off = 16; off > 0; off >>= 1) ss += __shfl_xor(ss, off, 32);
      float w = ss > 0.f ? ex / ss : 0.f;
      if (gv) s_w[(rb * 8 + q) * 20 + lane] = w;
    }
    // pooled: lane = output dim d
    float pq[8] = {0.f, 0.f, 0.f, 0.f, 0.f, 0.f, 0.f, 0.f};
    for (int g = 0; g < Gq; ++g) {
      float eg = s_enc[(rb * Gq + g) * 36 + lane];
#pragma unroll
      for (int q = 0; q < 8; ++q) pq[q] += s_w[(rb * 8 + q) * 20 + g] * eg;
    }
    // LayerNorm over 64 dims = [pq[2t](d) ; pq[2t+1](d+32)]
#pragma unroll
    for (int t = 0; t < 4; ++t) {
      float x0 = pq[2 * t], x1 = pq[2 * t + 1];
      float s = x0 + x1;
#pragma unroll
      for (int off = 16; off > 0; off >>= 1) s += __shfl_xor(s, off, 32);
      float mu = s * (1.f / 64.f);
      float d0 = x0 - mu, d1 = x1 - mu;
      float vv = d0 * d0 + d1 * d1;
#pragma unroll
      for (int off = 16; off > 0; off >>= 1) vv += __shfl_xor(vv, off, 32);
      float invn = rsqrtf(vv * (1.f / 64.f) + EPSq);
      s_hist[(rb * 4 + t) * 68 + lane] =
          d0 * invn * ln_g[t * 64 + lane] + ln_b[t * 64 + lane];
      s_hist[(rb * 4 + t) * 68 + 32 + lane] =
          d1 * invn * ln_g[t * 64 + 32 + lane] + ln_b[t * 64 + 32 + lane];
    }
  }
  __syncthreads();

  // ---- stage 2: backbone layer 1  [16,64] @ [64,256] (WMMA) ----
#pragma unroll
  for (int j = 0; j < 4; ++j) {
    int cb = (wv * 4 + j) * 16 + n;
    v8f acc = {0.f, 0.f, 0.f, 0.f, 0.f, 0.f, 0.f, 0.f};
#pragma unroll
    for (int s = 0; s < 16; ++s) {
      int kk = s * 4 + hi * 2;
      v2f a;  a.x = s_xs[n * 68 + kk];     a.y = s_xs[n * 68 + kk + 1];
      v2f bb; bb.x = W1[kk * H1q + cb];    bb.y = W1[(kk + 1) * H1q + cb];
      acc = wmma4(a, bb, acc);
    }
    float bi = b1[cb], mm = m1[cb], iv = rsqrtf(v1[cb] + EPSq), gg = g1[cb], bt = be1[cb];
#pragma unroll
    for (int r = 0; r < 8; ++r)
      s_z1[(r + hi * 8) * 260 + cb] = fmaxf((acc[r] + bi - mm) * iv * gg + bt, 0.f);
  }
  __syncthreads();

  // ---- stage 3: backbone layer 2  [16,256] @ [256,128] (WMMA) ----
#pragma unroll
  for (int j = 0; j < 2; ++j) {
    int cb = (wv * 2 + j) * 16 + n;
    v8f acc = {0.f, 0.f, 0.f, 0.f, 0.f, 0.f, 0.f, 0.f};
    for (int s = 0; s < 64; ++s) {
      int kk = s * 4 + hi * 2;
      v2f a;  a.x = s_z1[n * 260 + kk];    a.y = s_z1[n * 260 + kk + 1];
      v2f bb; bb.x = W2[kk * H2q + cb];    bb.y = W2[(kk + 1) * H2q + cb];
      acc = wmma4(a, bb, acc);
    }
    float bi = b2[cb], mm = m2[cb], iv = rsqrtf(v2[cb] + EPSq), gg = g2[cb], bt = be2[cb];
#pragma unroll
    for (int r = 0; r < 8; ++r)
      s_z2[(r + hi * 8) * 132 + cb] = fmaxf((acc[r] + bi - mm) * iv * gg + bt, 0.f);
  }
  __syncthreads();

  // ---- stage 4: per-target heads  [16,192] @ [192,32] (WMMA); wave wv owns t=wv
  const int t = wv;
  float rsum[8] = {0.f, 0.f, 0.f, 0.f, 0.f, 0.f, 0.f, 0.f};
#pragma unroll
  for (int j = 0; j < 2; ++j) {
    int cb = j * 16 + n;  // head hidden col 0..31
    v8f acc = {0.f, 0.f, 0.f, 0.f, 0.f, 0.f, 0.f, 0.f};
    for (int s = 0; s < 48; ++s) {
      int kk = s * 4 + hi * 2;  // even, so kk and kk+1 are on the same side of 128
      v2f a;
      if (kk < H2q) { a.x = s_z2[n * 132 + kk];            a.y = s_z2[n * 132 + kk + 1]; }
      else          { a.x = s_hist[(n * 4 + t) * 68 + kk - 128];
                      a.y = s_hist[(n * 4 + t) * 68 + kk - 127]; }
      v2f bb;
      bb.x = Wh1[(t * 192 + kk) * 32 + cb];
      bb.y = Wh1[(t * 192 + kk + 1) * 32 + cb];
      acc = wmma4(a, bb, acc);
    }
    float bh = bh1[t * 32 + cb], w2c = Wh2[t * 32 + cb];
#pragma unroll
    for (int r = 0; r < 8; ++r) {
      float p = fmaxf(acc[r] + bh, 0.f) * w2c;
#pragma unroll
      for (int off = 8; off > 0; off >>= 1) p += __shfl_xor(p, off, 16);
      rsum[r] += p;  // sum over this tile's 16 cols, rows r + 8*hi
    }
  }
  if (n == 0) {
    float bo = bh2[t];
#pragma unroll
    for (int r = 0; r < 8; ++r) {
      int rr = r + hi * 8;
      out[(long)(row0 + rr) * Tq + t] = fmaxf(rsum[r] + bo, 0.f);
    }
  }
}

// ---------------------------------------------------------------------------
extern "C" void kernel_launch(void* const* d_in, const int* in_sizes, int n_in,
                              void* d_out, int out_size, void* d_ws, size_t ws_size,
                              hipStream_t stream) {
  const float* x_static = (const float*)d_in[0];
  const float* x_kicks  = (const float*)d_in[1];
  const float* W_ke     = (const float*)d_in[2];
  const float* b_ke     = (const float*)d_in[3];
  const float* q_inner  = (const float*)d_in[4];
  const float* W_ge     = (const float*)d_in[5];
  const float* b_ge     = (const float*)d_in[6];
  const float* pos_emb  = (const float*)d_in[7];
  const float* q_outer  = (const float*)d_in[8];
  const float* log_temp = (const float*)d_in[9];
  const float* ln_g     = (const float*)d_in[10];
  const float* ln_b     = (const float*)d_in[11];
  const float* W1       = (const float*)d_in[12];
  const float* b1       = (const float*)d_in[13];
  const float* g1       = (const float*)d_in[14];
  const float* be1      = (const float*)d_in[15];
  const float* m1       = (const float*)d_in[16];
  const float* v1       = (const float*)d_in[17];
  const float* W2       = (const float*)d_in[18];
  const float* b2       = (const float*)d_in[19];
  const float* g2       = (const float*)d_in[20];
  const float* be2      = (const float*)d_in[21];
  const float* m2       = (const float*)d_in[22];
  const float* v2       = (const float*)d_in[23];
  const float* Wh1      = (const float*)d_in[24];
  const float* bh1      = (const float*)d_in[25];
  const float* Wh2      = (const float*)d_in[26];
  const float* bh2      = (const float*)d_in[27];
  const unsigned char* outer_mask = (const unsigned char*)d_in[28];
  const unsigned char* inner_mask = (const unsigned char*)d_in[29];

  float* enc_ws = (float*)d_ws;  // [B, G, DM] fp32 = ~34 MB

  const int games = Bq * Gq;                 // 278528, divisible by 8
  encode_games_kernel<<<games / 8, 256, 0, stream>>>(
      x_kicks, W_ke, b_ke, q_inner, W_ge, b_ge, pos_emb, inner_mask, enc_ws);

  fused_tail_kernel<<<Bq / 16, 128, 0, stream>>>(
      x_static, enc_ws, q_outer, log_temp, ln_g, ln_b,
      W1, b1, g1, be1, m1, v1, W2, b2, g2, be2, m2, v2,
      Wh1, bh1, Wh2, bh2, outer_mask, (float*)d_out);
}